// JointBilateralUpsampleLayer_38671885533456
// MI455X (gfx1250) — compile-verified
//
#include <hip/hip_runtime.h>

// ---------------------------------------------------------------------------
// Joint bilateral filter, WMMA-f32 formulation for gfx1250 (MI455X).
//
// out(16px x 21ch) = A(16 x 54) * S(54 x 21) per 16-pixel row segment, where
//   S  = flattened 3-row x 18-col src neighborhood (21 channels)
//   A  = banded matrix of per-pixel bilateral weights: A[m][dy*18+m+dx] = w
// Chained as 14 x V_WMMA_F32_16X16X4_F32 (K padded 54->56), 2 N-tiles (21->32).
// Memory-bound op (~94 MB total, fits in 192MB L2); WMMA keeps the weighted
// reduction off the VALU critical path.
// ---------------------------------------------------------------------------

typedef __attribute__((ext_vector_type(2))) float v2f;
typedef __attribute__((ext_vector_type(8))) float v8f;

#define B_  2
#define H_  512
#define W_  512
#define CS  21
#define CI  3
#define AK  56   // padded K (54 -> 56)
#define AST 57   // LDS row stride (odd -> conflict-free strided reads)

__device__ __forceinline__ int reflect_(int i, int n) {
    i = (i < 0) ? -i : i;                 // reflect (radius 1: single bounce)
    i = (i >= n) ? (2 * n - 2 - i) : i;
    return i;
}

__global__ __launch_bounds__(256) void jbu_wmma_kernel(
    const float* __restrict__ src, const float* __restrict__ im,
    float* __restrict__ out)
{
    __shared__ float Abuf[8][2][16][AST];  // per-wave, per-subtile A matrices

    const int wave = threadIdx.x >> 5;
    const int lane = threadIdx.x & 31;
    const int b  = blockIdx.z;
    const int y  = blockIdx.y;
    const int x0 = blockIdx.x * 256 + wave * 32;   // 32 pixels per wave

    const int sub_w = lane >> 4;   // subtile (0/1) of this lane's pixel
    const int m_w   = lane & 15;   // row within subtile

    // ---- per-lane pixel: 9 bilateral weights ------------------------------
    const int px_w = x0 + lane;
    const size_t rowc = ((size_t)(b * H_ + y)) * W_;
    const float c0 = im[(rowc + px_w) * CI + 0];
    const float c1 = im[(rowc + px_w) * CI + 1];
    const float c2 = im[(rowc + px_w) * CI + 2];

    const float W1[9] = { 0.36787944f, 0.60653066f, 0.36787944f,
                          0.60653066f, 1.00000000f, 0.60653066f,
                          0.36787944f, 0.60653066f, 0.36787944f };

    float w[9];
    float den = 0.0f;
#pragma unroll
    for (int i = 0; i < 9; ++i) {
        const int dy = i / 3 - 1, dx = i % 3 - 1;
        const int ry = reflect_(y + dy, H_);
        const int rx = reflect_(px_w + dx, W_);
        const float* p = im + (((size_t)(b * H_ + ry)) * W_ + rx) * CI;
        const float d0 = p[0] - c0, d1 = p[1] - c1, d2 = p[2] - c2;
        const float diff2 = d0 * d0 + d1 * d1 + d2 * d2;
        // exp(-diff2 / (2*0.25^2)) = exp(-8*diff2)
        w[i] = __expf(-diff2 * 8.0f) * W1[i];
        den += w[i];
    }
    const float inv_den = 1.0f / den;

    // ---- build banded A in LDS (each lane writes only its own row) --------
#pragma unroll
    for (int k = 0; k < AK; ++k) Abuf[wave][sub_w][m_w][k] = 0.0f;
#pragma unroll
    for (int i = 0; i < 9; ++i) {
        const int dy = i / 3, dx = i % 3;          // dx already dx_rel+1
        Abuf[wave][sub_w][m_w][dy * 18 + m_w + dx] = w[i];
    }
    __syncthreads();

    // reflected source rows for dy = -1, 0, +1
    const size_t rb0 = ((size_t)(b * H_ + reflect_(y - 1, H_))) * W_;
    const size_t rb1 = ((size_t)(b * H_ + y)) * W_;
    const size_t rb2 = ((size_t)(b * H_ + reflect_(y + 1, H_))) * W_;

    const int halfK = 2 * (lane >> 4);   // K sub-offset for this half-wave
    const int nlane = lane & 15;         // N (channel) lane / M row base

#pragma unroll
    for (int sub = 0; sub < 2; ++sub) {
        const int xbase = x0 + sub * 16 - 1;   // neighborhood col 0 -> x-1
#pragma unroll
        for (int nt = 0; nt < 2; ++nt) {
            const int c     = nt * 16 + nlane;
            const int c_eff = (c < CS) ? c : (CS - 1);   // clamp pad channels

            v8f acc = {};
#pragma unroll
            for (int t = 0; t < 14; ++t) {
                const int kb = 4 * t + halfK;

                // A fragment: 16x4 f32 layout (lane=M, VGPR0/1 = K pair)
                v2f a;
                a.x = Abuf[wave][sub][nlane][kb];
                a.y = Abuf[wave][sub][nlane][kb + 1];

                // B fragment: 4x16 f32 layout (lane=N, VGPR0/1 = K pair)
                v2f bb;
#pragma unroll
                for (int d = 0; d < 2; ++d) {
                    const int kk   = kb + d;
                    const int sidx = (kk < 54) ? kk : 53;      // K pad -> safe
                    const int dy   = (sidx >= 36) ? 2 : ((sidx >= 18) ? 1 : 0);
                    const int col  = sidx - dy * 18;
                    const int rx   = reflect_(xbase + col, W_);
                    const size_t rb = (dy == 0) ? rb0 : ((dy == 1) ? rb1 : rb2);
                    const float v = src[(rb + rx) * CS + c_eff];
                    if (d == 0) bb.x = v; else bb.y = v;
                }

                acc = __builtin_amdgcn_wmma_f32_16x16x4_f32(
                        /*neg_a=*/false, a, /*neg_b=*/false, bb,
                        /*c_mod=*/(short)0, acc,
                        /*reuse_a=*/false, /*reuse_b=*/false);
            }

            // ---- normalize + store (C/D layout: VGPR j -> M=j(+8)) --------
#pragma unroll
            for (int j = 0; j < 8; ++j) {
                const int M       = j + ((lane >> 4) << 3);
                const int srcLane = sub * 16 + M;        // lane holding 1/den
                const float invd  = __shfl(inv_den, srcLane, 32);
                const int pxo     = x0 + sub * 16 + M;
                const float v     = acc[j] * invd;
                if (c < CS)
                    out[(rowc + pxo) * CS + c] = v;
            }
        }
    }
}

extern "C" void kernel_launch(void* const* d_in, const int* in_sizes, int n_in,
                              void* d_out, int out_size, void* d_ws, size_t ws_size,
                              hipStream_t stream) {
    (void)in_sizes; (void)n_in; (void)d_ws; (void)ws_size; (void)out_size;
    const float* src = (const float*)d_in[0];   // [2,512,512,21]
    const float* im  = (const float*)d_in[1];   // [2,512,512,3]
    float* out = (float*)d_out;                 // [2,512,512,21]

    dim3 grid(W_ / 256, H_, B_);   // (2, 512, 2)
    dim3 block(256);               // 8 waves; each wave = 32 pixels of a row
    jbu_wmma_kernel<<<grid, block, 0, stream>>>(src, im, out);
}